// RandomShiftsAug_1168231104879
// MI455X (gfx1250) — compile-verified
//
#include <hip/hip_runtime.h>
#include <cstdint>

// RandomShiftsAug: out[n,c,i,j] = x[n,c, clamp(i+sy-4,0,83), clamp(j+sx-4,0,83)]
// sx = shift[n,0], sy = shift[n,1]. Pure bandwidth-bound gather (~260MB total,
// ~11us at 23.3 TB/s on MI455X). Data path: CDNA5 async global->LDS copy engine
// (ASYNCcnt) for the clamped row gather, double-buffered LDS, software-pipelined
// 4 rows/wave with partial s_wait_asynccnt, aligned b128 stores on the way out.

namespace {
constexpr int kN = 512;
constexpr int kC = 9;
constexpr int kH = 84;
constexpr int kW = 84;            // 84 floats = 336 B = 21 * 16 B -> rows 16B aligned
constexpr int kPad = 4;
constexpr int kRows = kN * kC * kH;   // 387072 output rows
constexpr int kWavesPerBlock = 8;     // 256 threads, wave32
constexpr int kRowsPerWave = 4;       // 387072 = 96768 waves * 4 rows exactly
constexpr int kRowBufFloats = 96;     // 3 chunks of 32 lanes (84 used)
}

struct RowParams {
    unsigned srcRowBase;  // element index of clamped source row start
    unsigned dstRowBase;  // element index of output row start
    int sxm4;             // sx - pad
};

__device__ __forceinline__ RowParams row_params(int row, const int* __restrict__ shift) {
    const int n   = row / (kC * kH);
    const int rem = row - n * (kC * kH);
    const int c   = rem / kH;
    const int i   = rem - c * kH;

    const int sx = shift[2 * n + 0];
    const int sy = shift[2 * n + 1];

    int isrc = i + sy - kPad;
    isrc = isrc < 0 ? 0 : (isrc > kH - 1 ? kH - 1 : isrc);

    RowParams p;
    p.srcRowBase = (unsigned)(((n * kC + c) * kH + isrc) * kW);
    p.dstRowBase = (unsigned)(row * kW);
    p.sxm4       = sx - kPad;
    return p;
}

// Issue 3x32 async b32 gathers for one row into the LDS buffer at ldsBase.
// All 32 lanes active (EXEC all ones); slots >= 84 clamp to col 83, never read.
__device__ __forceinline__ void issue_row_loads(const float* __restrict__ x,
                                                const RowParams& p,
                                                unsigned ldsBase, int lane) {
#pragma unroll
    for (int k = 0; k < 3; ++k) {
        const int e   = k * 32 + lane;
        const int col = (e < kW) ? e : (kW - 1);
        int jsrc = col + p.sxm4;
        jsrc = jsrc < 0 ? 0 : (jsrc > kW - 1 ? kW - 1 : jsrc);

        const unsigned goff  = (p.srcRowBase + (unsigned)jsrc) * 4u;  // byte offset
        const unsigned laddr = ldsBase + (unsigned)e * 4u;            // LDS byte addr

        // GVS mode: loads the dword at base x (SGPR pair) plus byte offset goff into the LDS address in laddr
        asm volatile("global_load_async_to_lds_b32 %0, %1, %2"
                     :
                     : "v"(laddr), "v"(goff), "s"(x)
                     : "memory");
    }
}

__global__ __launch_bounds__(256)
void shift_aug_async_kernel(const float* __restrict__ x,
                            const int* __restrict__ shift,
                            float* __restrict__ out)
{
    __shared__ __align__(16) float smem[kWavesPerBlock][2][kRowBufFloats];

    const int lane = threadIdx.x & 31;
    const int wave = threadIdx.x >> 5;
    const int waveGlobal = blockIdx.x * kWavesPerBlock + wave;
    const int r0 = waveGlobal * kRowsPerWave;
    if (r0 >= kRows) return;

    unsigned ldsBase[2];
    ldsBase[0] = (unsigned)(uintptr_t)(&smem[wave][0][0]);
    ldsBase[1] = (unsigned)(uintptr_t)(&smem[wave][1][0]);

    // Prologue: fill buffer 0 with row r0.
    RowParams cur = row_params(r0, shift);
    issue_row_loads(x, cur, ldsBase[0], lane);

#pragma unroll
    for (int k = 0; k < kRowsPerWave; ++k) {
        const unsigned dstRowBase = cur.dstRowBase;

        if (k + 1 < kRowsPerWave) {
            // Issue next row's gathers into the other buffer, then wait for
            // only the current row's 3 loads (async loads complete in order;
            // ASYNCcnt holds loads only -- stores go through STOREcnt).
            RowParams nxt = row_params(r0 + k + 1, shift);
            issue_row_loads(x, nxt, ldsBase[(k + 1) & 1], lane);
            cur = nxt;
            asm volatile("s_wait_asynccnt 0x3" ::: "memory");
        } else {
            asm volatile("s_wait_asynccnt 0x0" ::: "memory");
        }

        // Drain current row: ds_load_b128 + global_store_b128 (21 lanes).
        // Row byte base = row*336, a multiple of 16 -> aligned 16B stores.
        // The store depends on the ds_load result, so dscnt is drained before
        // the next iteration's async issue can overwrite this buffer.
        if (lane < kW / 4) {
            const float4 v = *reinterpret_cast<const float4*>(
                &smem[wave][k & 1][lane * 4]);
            *reinterpret_cast<float4*>(out + dstRowBase + (unsigned)(lane * 4)) = v;
        }
    }
}

extern "C" void kernel_launch(void* const* d_in, const int* in_sizes, int n_in,
                              void* d_out, int out_size, void* d_ws, size_t ws_size,
                              hipStream_t stream)
{
    (void)in_sizes; (void)n_in; (void)out_size; (void)d_ws; (void)ws_size;

    const float* x     = (const float*)d_in[0];
    const int*   shift = (const int*)d_in[1];
    float*       out   = (float*)d_out;

    const int waves  = (kRows + kRowsPerWave - 1) / kRowsPerWave;        // 96768
    const int blocks = (waves + kWavesPerBlock - 1) / kWavesPerBlock;    // 12096
    hipLaunchKernelGGL(shift_aug_async_kernel,
                       dim3(blocks), dim3(kWavesPerBlock * 32), 0, stream,
                       x, shift, out);
}